// Net_50440095924334
// MI455X (gfx1250) — compile-verified
//
#include <hip/hip_runtime.h>
#include <hip/hip_bf16.h>

// ---------------------------------------------------------------------------
// Problem constants (match reference)
// ---------------------------------------------------------------------------
#define N_TOTAL     2048
#define S_SRC       1024
#define T_TGT       1024
#define FEAT_IN     25088
#define D_FC        512
#define D_G1        512
#define D_G2        256
#define NUM_CLASSES 4

typedef __attribute__((ext_vector_type(2))) float v2f;
typedef __attribute__((ext_vector_type(8))) float v8f;

// Generic pointer -> raw LDS byte offset (VDST operand of async-to-LDS ops is
// a byte offset from the wave's LDS base, ISA §15.18).
__device__ static inline unsigned lds_off(const void* p)
{
    return (unsigned)(unsigned long long)
        (const __attribute__((address_space(3))) char*)p;
}

// ---------------------------------------------------------------------------
// Generic fp32 WMMA GEMM:  C[M,N] = op(A[M,K] @ B[K,N] (+ bias)),  row-major.
// Requirements (all call sites satisfy them): M%32==0, N%256==0, K%32==0.
// Block = 256 threads = 8 wave32s, tile BM=32 x BN=256, K-tile KT=32.
// Wave grid 2x4; each wave computes a 16x64 sub-tile = 4 WMMA C tiles.
//
// Staging:
//   A tile : GLOBAL_LOAD_ASYNC_TO_LDS_B128 (ASYNCcnt path) -- each thread
//            DMA-copies one contiguous 16B chunk straight into As (layout-
//            preserving stream, no register transform needed).
//   B tile : global->VGPR->LDS with k-pair interleave (the swizzle an async
//            straight-copy cannot do).
//
// LDS layouts:
//   As[BM][KT+4]      : A fragment (k,k+1) pair -> one ds_load_b64,
//                       stride 36 words -> conflict-free across the wave.
//   Bs[KT/2][2*BN+32] : k-pair interleaved: (k,n) at kp*LDB + 2n + (k&1).
//                       B fragment {B[k][c],B[k+1][c]} -> one aligned
//                       ds_load_b64 into an adjacent VGPR pair (no movs).
//                       LDB = 544 = 32 mod 64 banks -> the two half-wave
//                       phases (rows kp / kp+1) hit disjoint bank halves.
//
// Fragment layouts per CDNA5 ISA 7.12.2:
//   A 16x4 f32 : lanes 0-15 -> M=lane, VGPR0/1 = K0/K1 ; lanes 16-31 -> K2/K3
//   B 4x16 f32 : lanes 0-15 -> N=lane, VGPR0/1 = K0/K1 ; lanes 16-31 -> K2/K3
//   C 16x16    : VGPR r: lanes 0-15 -> (M=r, N=lane); lanes 16-31 -> (M=r+8)
// ---------------------------------------------------------------------------
template <bool BIAS, bool RELU>
__global__ __launch_bounds__(256) void gemm_wmma_f32(
    const float* __restrict__ A, const float* __restrict__ B,
    const float* __restrict__ bias, float* __restrict__ C,
    int M, int N, int K)
{
    constexpr int BM  = 32, BN = 256, KT = 32;
    constexpr int LDA = KT + 4;                 // 36 words
    constexpr int LDB = 2 * BN + 32;            // 544 words per k-pair row
    __shared__ float As[BM * LDA];              // 4.5 KB
    __shared__ float Bs[(KT / 2) * LDB];        // 34 KB

    const int tid  = threadIdx.x;
    const int lane = tid & 31;
    const int wave = tid >> 5;
    const int wrow = (wave >> 2) << 4;          // 0 or 16
    const int wcol = (wave & 3) << 6;           // 0,64,128,192
    const int row0 = blockIdx.x * BM;
    const int col0 = blockIdx.y * BN;

    // Cooperative staging indices.
    const int ar = tid >> 3;                    // A row within tile  (0..31)
    const int ak = (tid & 7) << 2;              // A k within tile    (0..28)
    const unsigned as_dst = lds_off(As + ar * LDA + ak);   // 16B-aligned

    v8f acc[4] = {};

    for (int k0 = 0; k0 < K; k0 += KT) {
        // --- B tile: global -> registers (overlaps prior compute) ----------
        float4 b0[4], b1[4];
#pragma unroll
        for (int r = 0; r < 4; ++r) {
            int u  = (r << 8) + tid;            // 0..1023 (kpair x n-quad unit)
            int kp = u >> 6;                    // 0..15
            int n  = (u & 63) << 2;             // 0..252
            const float* gp = B + (size_t)(k0 + (kp << 1)) * N + col0 + n;
            b0[r] = *(const float4*)gp;         // row 2*kp   (coalesced)
            b1[r] = *(const float4*)(gp + N);   // row 2*kp+1 (coalesced)
        }
        // prefetch next A tile (speculative; OOB prefetch is dropped by HW)
        __builtin_prefetch(A + (size_t)(row0 + ar) * K + k0 + KT + ak, 0, 1);

        __syncthreads();                        // prior tile fully consumed

        // --- A tile: async DMA global -> LDS (ASYNCcnt) --------------------
        {
            const float* agp = A + (size_t)(row0 + ar) * K + k0 + ak;
            asm volatile("global_load_async_to_lds_b128 %0, %1, off"
                         :: "v"(as_dst),
                            "v"((unsigned long long)agp)
                         : "memory");
        }

        // --- B tile: interleaved store to LDS ------------------------------
#pragma unroll
        for (int r = 0; r < 4; ++r) {
            int u  = (r << 8) + tid;
            int kp = u >> 6;
            int n  = (u & 63) << 2;
            float2* bp = (float2*)(Bs + kp * LDB + (n << 1)); // 32B contiguous
            bp[0] = make_float2(b0[r].x, b1[r].x);
            bp[1] = make_float2(b0[r].y, b1[r].y);
            bp[2] = make_float2(b0[r].z, b1[r].z);
            bp[3] = make_float2(b0[r].w, b1[r].w);
        }
        asm volatile("s_wait_asynccnt 0x0" ::: "memory");   // A DMA landed
        __syncthreads();

        // --- compute 8 k-steps of 4 ---------------------------------------
        const int m  = lane & 15;
        const int ko = (lane >> 4) << 1;        // 0 or 2
#pragma unroll
        for (int kk = 0; kk < KT; kk += 4) {
            const int kf = kk + ko;             // even
            float2 a2 = *(const float2*)(As + (wrow + m) * LDA + kf);
            v2f af = { a2.x, a2.y };
#pragma unroll
            for (int nt = 0; nt < 4; ++nt) {
                int c = wcol + (nt << 4) + (lane & 15);
                float2 b2 = *(const float2*)(Bs + (kf >> 1) * LDB + (c << 1));
                v2f bf = { b2.x, b2.y };
                acc[nt] = __builtin_amdgcn_wmma_f32_16x16x4_f32(
                    false, af, false, bf, (short)0, acc[nt], false, false);
            }
        }
    }

    // --- epilogue ----------------------------------------------------------
    const int mbase = row0 + wrow + ((lane >> 4) << 3);
#pragma unroll
    for (int nt = 0; nt < 4; ++nt) {
        int n = col0 + wcol + (nt << 4) + (lane & 15);
        float bb = BIAS ? bias[n] : 0.0f;
#pragma unroll
        for (int r = 0; r < 8; ++r) {
            float v = acc[nt][r] + bb;
            if (RELU) v = fmaxf(v, 0.0f);
            C[(size_t)(mbase + r) * N + n] = v;
        }
    }
}

// ---------------------------------------------------------------------------
// centers[c][:] = segment_sum(feats[:S], labels) ; one thread per column
// ---------------------------------------------------------------------------
__global__ __launch_bounds__(256) void centers_kernel(
    const float* __restrict__ feats, const int* __restrict__ labels,
    float* __restrict__ centers)
{
    int c = blockIdx.x * blockDim.x + threadIdx.x;   // 0..511
    float a0 = 0.f, a1 = 0.f, a2 = 0.f, a3 = 0.f;
    for (int r = 0; r < S_SRC; ++r) {
        float v = feats[(size_t)r * D_FC + c];
        int   l = labels[r];
        a0 += (l == 0) ? v : 0.f;
        a1 += (l == 1) ? v : 0.f;
        a2 += (l == 2) ? v : 0.f;
        a3 += (l == 3) ? v : 0.f;
    }
    centers[0 * D_FC + c] = a0;
    centers[1 * D_FC + c] = a1;
    centers[2 * D_FC + c] = a2;
    centers[3 * D_FC + c] = a3;
}

__global__ void zero_counts_kernel(int* cnt)
{
    if (threadIdx.x < 2 * NUM_CLASSES) cnt[threadIdx.x] = 0;
}

__global__ __launch_bounds__(256) void src_labels_kernel(
    const int* __restrict__ src, int* __restrict__ all_labels, int* cnt)
{
    int i = blockIdx.x * blockDim.x + threadIdx.x;
    if (i < S_SRC) {
        int l = src[i];
        all_labels[i] = l;
        atomicAdd(&cnt[l], 1);
    }
}

// ---------------------------------------------------------------------------
// Per target row: argmin_j (||c_j||^2 - 2 t.c_j)  -- one wave32 per row.
// ---------------------------------------------------------------------------
__global__ __launch_bounds__(256) void assign_kernel(
    const float* __restrict__ feats, const float* __restrict__ centers,
    int* __restrict__ all_labels, int* __restrict__ cnt)
{
    int wave = (blockIdx.x * blockDim.x + threadIdx.x) >> 5;
    int lane = threadIdx.x & 31;
    if (wave >= T_TGT) return;

    const float* t = feats + (size_t)(S_SRC + wave) * D_FC;
    float dot[NUM_CLASSES] = {0.f, 0.f, 0.f, 0.f};
    float csq[NUM_CLASSES] = {0.f, 0.f, 0.f, 0.f};
    for (int c = lane; c < D_FC; c += 32) {
        float v = t[c];
#pragma unroll
        for (int j = 0; j < NUM_CLASSES; ++j) {
            float cv = centers[j * D_FC + c];
            dot[j] += v * cv;
            csq[j] += cv * cv;
        }
    }
#pragma unroll
    for (int j = 0; j < NUM_CLASSES; ++j) {
#pragma unroll
        for (int off = 16; off > 0; off >>= 1) {
            dot[j] += __shfl_xor(dot[j], off, 32);
            csq[j] += __shfl_xor(csq[j], off, 32);
        }
    }
    if (lane == 0) {
        float best = csq[0] - 2.f * dot[0];
        int   bj   = 0;
#pragma unroll
        for (int j = 1; j < NUM_CLASSES; ++j) {
            float s = csq[j] - 2.f * dot[j];
            if (s < best) { best = s; bj = j; }   // first-min tiebreak
        }
        all_labels[S_SRC + wave] = bj;
        atomicAdd(&cnt[NUM_CLASSES + bj], 1);
    }
}

// deg(i<S)  = 2 + #tgt(label)
// deg(i>=S) = 1 + #src(label) + #tgt(label)
__global__ __launch_bounds__(256) void dinv_kernel(
    const int* __restrict__ all_labels, const int* __restrict__ cnt,
    float* __restrict__ dinv)
{
    int i = blockIdx.x * blockDim.x + threadIdx.x;
    if (i >= N_TOTAL) return;
    int   l   = all_labels[i];
    float deg = (i < S_SRC) ? (2.0f + (float)cnt[NUM_CLASSES + l])
                            : (1.0f + (float)cnt[l] + (float)cnt[NUM_CLASSES + l]);
    dinv[i] = rsqrtf(deg);
}

// A_norm[i][j] = dinv_i * dinv_j * a ; a = 2 on diag, 0 in src-src block,
// else 1{label_i==label_j}. 4 elements per thread, vectorized store.
__global__ __launch_bounds__(256) void anorm_kernel(
    const int* __restrict__ labels, const float* __restrict__ dinv,
    float* __restrict__ Aout)
{
    int e  = (blockIdx.x * blockDim.x + threadIdx.x) << 2;
    int i  = e >> 11;                // /2048
    int j0 = e & (N_TOTAL - 1);
    int   li = labels[i];
    float di = dinv[i];
    float o[4];
#pragma unroll
    for (int q = 0; q < 4; ++q) {
        int j  = j0 + q;
        int lj = labels[j];
        float a;
        if (i == j)                      a = 2.0f;
        else if (i < S_SRC && j < S_SRC) a = 0.0f;
        else                             a = (li == lj) ? 1.0f : 0.0f;
        o[q] = a * di * dinv[j];
    }
    *(float4*)(Aout + (size_t)i * N_TOTAL + j0) = make_float4(o[0], o[1], o[2], o[3]);
}

// ---------------------------------------------------------------------------
// Launch: full pipeline
//   d_out = [ h (2048x256) | A_norm (2048x2048) | feats (2048x512) ]
// ---------------------------------------------------------------------------
extern "C" void kernel_launch(void* const* d_in, const int* in_sizes, int n_in,
                              void* d_out, int out_size, void* d_ws, size_t ws_size,
                              hipStream_t stream)
{
    const float* x    = (const float*)d_in[0];
    const float* Wfc  = (const float*)d_in[1];
    const float* bfc  = (const float*)d_in[2];
    const float* Wg1  = (const float*)d_in[3];
    const float* bg1  = (const float*)d_in[4];
    const float* Wg2  = (const float*)d_in[5];
    const float* bg2  = (const float*)d_in[6];
    const int*   slab = (const int*)d_in[7];
    (void)in_sizes; (void)n_in; (void)out_size; (void)ws_size;

    float* out_h     = (float*)d_out;
    float* out_anorm = out_h + (size_t)N_TOTAL * D_G2;
    float* out_feats = out_anorm + (size_t)N_TOTAL * N_TOTAL;

    float* centers = (float*)d_ws;                         // 4*512 f
    int*   cnt     = (int*)(centers + NUM_CLASSES * D_FC); // 8 i
    int*   labels  = cnt + 2 * NUM_CLASSES;                // 2048 i
    float* dinv    = (float*)(labels + N_TOTAL);           // 2048 f
    float* t1      = dinv + N_TOTAL;                       // 2048*512 f
    float* h1      = t1 + (size_t)N_TOTAL * D_G1;          // 2048*512 f
    float* t2      = h1 + (size_t)N_TOTAL * D_G1;          // 2048*256 f

    // 1) feats = x @ W_fc + b_fc   (the 52.7 GFLOP GEMM)
    gemm_wmma_f32<true, false><<<dim3(N_TOTAL / 32, D_FC / 256), 256, 0, stream>>>(
        x, Wfc, bfc, out_feats, N_TOTAL, D_FC, FEAT_IN);

    // 2) class centers + labels + degrees + A_norm
    zero_counts_kernel<<<1, 32, 0, stream>>>(cnt);
    centers_kernel<<<D_FC / 256, 256, 0, stream>>>(out_feats, slab, centers);
    src_labels_kernel<<<S_SRC / 256, 256, 0, stream>>>(slab, labels, cnt);
    assign_kernel<<<T_TGT / 8, 256, 0, stream>>>(out_feats, centers, labels, cnt);
    dinv_kernel<<<N_TOTAL / 256, 256, 0, stream>>>(labels, cnt, dinv);
    anorm_kernel<<<(N_TOTAL * (N_TOTAL / 4)) / 256, 256, 0, stream>>>(
        labels, dinv, out_anorm);

    // 3) GCN layer 1: h1 = relu(A_norm @ (feats @ W_g1) + b_g1)
    gemm_wmma_f32<false, false><<<dim3(N_TOTAL / 32, D_G1 / 256), 256, 0, stream>>>(
        out_feats, Wg1, nullptr, t1, N_TOTAL, D_G1, D_FC);
    gemm_wmma_f32<true, true><<<dim3(N_TOTAL / 32, D_G1 / 256), 256, 0, stream>>>(
        out_anorm, t1, bg1, h1, N_TOTAL, D_G1, N_TOTAL);

    // 4) GCN layer 2: h = relu(A_norm @ (h1 @ W_g2) + b_g2)
    gemm_wmma_f32<false, false><<<dim3(N_TOTAL / 32, D_G2 / 256), 256, 0, stream>>>(
        h1, Wg2, nullptr, t2, N_TOTAL, D_G2, D_G1);
    gemm_wmma_f32<true, true><<<dim3(N_TOTAL / 32, D_G2 / 256), 256, 0, stream>>>(
        out_anorm, t2, bg2, out_h, N_TOTAL, D_G2, N_TOTAL);
}